// DiversityMetric_7447473291846
// MI455X (gfx1250) — compile-verified
//
#include <hip/hip_runtime.h>
#include <math.h>

typedef __attribute__((ext_vector_type(2))) float v2f;
typedef __attribute__((ext_vector_type(4))) float v4f;
typedef __attribute__((ext_vector_type(8))) float v8f;

#define B_SZ   16
#define N_PTS  2048
#define D_DIM  128

// ---------------------------------------------------------------------------
// Kernel 1: squared norms of every row:  sn[b*N + n] = sum_k X[b,n,k]^2
// ---------------------------------------------------------------------------
__global__ void __launch_bounds__(256)
div_snorm_kernel(const float* __restrict__ X, float* __restrict__ sn) {
    const int row = blockIdx.x * blockDim.x + threadIdx.x;   // 0 .. B*N-1
    const float* p = X + (size_t)row * D_DIM;
    float s = 0.0f;
#pragma unroll
    for (int k = 0; k < D_DIM; k += 4) {
        v4f v = *(const v4f*)(p + k);
        s += v.x * v.x + v.y * v.y + v.z * v.z + v.w * v.w;
    }
    sn[row] = s;
}

// ---------------------------------------------------------------------------
// Kernel 2: nearest-neighbor distance per point via WMMA fp32 Gram tiles.
//
// Grid: B * (N/64) workgroups, 128 threads (4 waves). Wave w owns the 16-row
// A-tile starting at i0 = rowblock*64 + w*16 and scans all 128 column tiles.
//
// V_WMMA_F32_16X16X4_F32 lane layout (wave32):
//   A (16x4, MxK):  lane L, vgpr v -> M = L%16, K = 2*(L/16) + v
//   B (4x16, KxN):  lane L, vgpr v -> N = L%16, K = 2*(L/16) + v
//   C/D (16x16):    vgpr r, lane L -> M = r + 8*(L/16), N = L%16
// Since both A and B index their point by (L%16) and K by (2*(L/16)+v), both
// tiles are plain contiguous float2 loads from row-major X.
// ---------------------------------------------------------------------------
__global__ void __launch_bounds__(128)
div_nnd_kernel(const float* __restrict__ X, const float* __restrict__ sn,
               float* __restrict__ nnd) {
    const int b    = blockIdx.x >> 5;        // / (N/64 = 32)
    const int rb   = blockIdx.x & 31;
    const int wave = threadIdx.x >> 5;
    const int lane = threadIdx.x & 31;
    const int i0   = rb * 64 + wave * 16;
    const int nlo  = lane & 15;
    const int half = lane >> 4;
    const int kh   = half * 2;

    const float* Xb  = X  + (size_t)b * N_PTS * D_DIM;
    const float* snb = sn + b * N_PTS;

    // Preload this wave's A tile (rows i0..i0+15, K=0..127) in WMMA A layout.
    v2f a[32];
    const float* arow = Xb + (size_t)(i0 + nlo) * D_DIM + kh;
#pragma unroll
    for (int kk = 0; kk < 32; ++kk)
        a[kk] = *(const v2f*)(arow + kk * 4);

    float sni[8];
#pragma unroll
    for (int r = 0; r < 8; ++r)
        sni[r] = snb[i0 + r + 8 * half];

    float minv[8];
#pragma unroll
    for (int r = 0; r < 8; ++r)
        minv[r] = __builtin_inff();

    for (int jt = 0; jt < N_PTS / 16; jt += 2) {
        const int j0 = jt * 16;
        const int j1 = j0 + 16;
        v8f acc0 = {};
        v8f acc1 = {};
        const float* b0 = Xb + (size_t)(j0 + nlo) * D_DIM + kh;
        const float* b1 = Xb + (size_t)(j1 + nlo) * D_DIM + kh;
#pragma unroll
        for (int kk = 0; kk < 32; ++kk) {
            v2f bv0 = *(const v2f*)(b0 + kk * 4);
            v2f bv1 = *(const v2f*)(b1 + kk * 4);
            acc0 = __builtin_amdgcn_wmma_f32_16x16x4_f32(
                false, a[kk], false, bv0, (short)0, acc0, false, false);
            acc1 = __builtin_amdgcn_wmma_f32_16x16x4_f32(
                false, a[kk], false, bv1, (short)0, acc1, false, false);
        }
        const float snj0 = snb[j0 + nlo];
        const float snj1 = snb[j1 + nlo];
#pragma unroll
        for (int r = 0; r < 8; ++r) {
            const int gi = i0 + r + 8 * half;
            float sq0 = fmaxf(sni[r] + snj0 - 2.0f * acc0[r], 0.0f);
            float sq1 = fmaxf(sni[r] + snj1 - 2.0f * acc1[r], 0.0f);
            if (gi != (j0 + nlo)) minv[r] = fminf(minv[r], sq0);
            if (gi != (j1 + nlo)) minv[r] = fminf(minv[r], sq1);
        }
    }

    // Min-reduce each row across the 16 N-lanes of its half-wave; lane n==0
    // of each half writes sqrt(min squared distance).
#pragma unroll
    for (int r = 0; r < 8; ++r) {
        float v = minv[r];
#pragma unroll
        for (int m = 1; m < 16; m <<= 1)
            v = fminf(v, __shfl_xor(v, m, 32));
        if (nlo == 0)
            nnd[b * N_PTS + i0 + r + 8 * half] = sqrtf(v);
    }
}

// ---------------------------------------------------------------------------
// Kernel 3: mean / unbiased std / CV over all B*N nnd values (f64 accum).
// ---------------------------------------------------------------------------
__global__ void __launch_bounds__(1024)
div_stats_kernel(const float* __restrict__ nnd, float* __restrict__ out) {
    const int tid = threadIdx.x;
    double s = 0.0, s2 = 0.0;
    for (int i = tid; i < B_SZ * N_PTS; i += 1024) {
        const double v = (double)nnd[i];
        s  += v;
        s2 += v * v;
    }
    for (int m = 16; m >= 1; m >>= 1) {
        s  += __shfl_xor(s,  m, 32);
        s2 += __shfl_xor(s2, m, 32);
    }
    __shared__ double ls[32], ls2[32];
    const int wv = tid >> 5, ln = tid & 31;
    if (ln == 0) { ls[wv] = s; ls2[wv] = s2; }
    __syncthreads();
    if (tid == 0) {
        double S = 0.0, S2 = 0.0;
        for (int w = 0; w < 32; ++w) { S += ls[w]; S2 += ls2[w]; }
        const double n    = (double)(B_SZ * N_PTS);
        const double mean = S / n;
        double var = (S2 - S * S / n) / (n - 1.0);
        if (var < 0.0) var = 0.0;
        const double stdd = sqrt(var);
        const double cv   = (mean > 1e-8) ? stdd / fmax(mean, 1e-8) : 0.0;
        out[0] = (float)mean;
        out[1] = (float)stdd;
        out[2] = (float)cv;
    }
}

// ---------------------------------------------------------------------------
extern "C" void kernel_launch(void* const* d_in, const int* in_sizes, int n_in,
                              void* d_out, int out_size, void* d_ws, size_t ws_size,
                              hipStream_t stream) {
    (void)in_sizes; (void)n_in; (void)out_size; (void)ws_size;
    const float* X   = (const float*)d_in[0];
    float*       out = (float*)d_out;
    float*       sn  = (float*)d_ws;                 // B*N floats (128 KB)
    float*       nnd = sn + B_SZ * N_PTS;            // B*N floats (128 KB)

    div_snorm_kernel<<<(B_SZ * N_PTS) / 256, 256, 0, stream>>>(X, sn);
    div_nnd_kernel<<<B_SZ * (N_PTS / 64), 128, 0, stream>>>(X, sn, nnd);
    div_stats_kernel<<<1, 1024, 0, stream>>>(nnd, out);
}